// EdgePredictorDGL_15925738734016
// MI455X (gfx1250) — compile-verified
//
#include <hip/hip_runtime.h>
#include <hip/hip_bf16.h>

typedef __attribute__((ext_vector_type(2))) float v2f;
typedef __attribute__((ext_vector_type(8))) float v8f;

#define D 128          // D_IN == D_HID == 128
#define WAVE 32

// ---------- order-preserving float<->uint key (for segment max via atomicMax) ----------
__device__ __forceinline__ unsigned f2key(float f) {
  unsigned b = __float_as_uint(f);
  return (b & 0x80000000u) ? ~b : (b | 0x80000000u);
}
__device__ __forceinline__ float key2f(unsigned k) {
  unsigned b = (k & 0x80000000u) ? (k ^ 0x80000000u) : ~k;
  return __uint_as_float(b);
}

// ---------- 1) degree counts ----------
__global__ void k_degrees(const int* __restrict__ src, const int* __restrict__ dst,
                          float* __restrict__ deg_out, float* __restrict__ deg_in, int E) {
  int i = blockIdx.x * blockDim.x + threadIdx.x;
  if (i < E) {
    atomicAdd(&deg_out[src[i]], 1.0f);
    atomicAdd(&deg_in[dst[i]], 1.0f);
  }
}

// ---------- 2) agg[dst] += x[src] * rsqrt(max(deg_out[src],1)) ; one wave per edge ----------
__global__ void k_aggregate(const float* __restrict__ x, const int* __restrict__ src,
                            const int* __restrict__ dst, const float* __restrict__ deg_out,
                            float* __restrict__ agg, int E) {
  int lane = threadIdx.x & (WAVE - 1);
  int e = (blockIdx.x * blockDim.x + threadIdx.x) >> 5;
  if (e >= E) return;
  int s = src[e], d = dst[e];
  float sc = rsqrtf(fmaxf(deg_out[s], 1.0f));
  float4 v = ((const float4*)(x + (size_t)s * D))[lane];   // 32 lanes * 4 = 128 channels
  float* ag = agg + (size_t)d * D + lane * 4;
  atomicAdd(ag + 0, v.x * sc);
  atomicAdd(ag + 1, v.y * sc);
  atomicAdd(ag + 2, v.z * sc);
  atomicAdd(ag + 3, v.w * sc);
}

// ---------- one k-step: 8 WMMAs sharing one A fragment ----------
__device__ __forceinline__ void wmma_kstep(v8f acc[8], v2f a_raw, float scale,
                                           const float* __restrict__ W_lds,
                                           int k0, int half, int l) {
  v2f a;
  a.x = a_raw.x * scale;
  a.y = a_raw.y * scale;
  const float* w0 = &W_lds[(k0 + half * 2) * D + l];
  #pragma unroll
  for (int tn = 0; tn < 8; ++tn) {
    v2f b;
    b.x = w0[tn * 16];
    b.y = w0[tn * 16 + D];
    acc[tn] = __builtin_amdgcn_wmma_f32_16x16x4_f32(
        false, a, false, b, (short)0, acc[tn], false, false);
  }
}

// ---------- 3) WMMA GEMM: C[M,128] = (A * rowscale) @ W[128,128] + bias ----------
// W staged once per block into LDS via CDNA5 async global->LDS (ASYNCcnt), then
// 8 waves feed v_wmma_f32_16x16x4_f32 from LDS.  One wave = 16x128 row strip
// (8 v8f accumulators).  The k-loop is manually unrolled x2 with two named A
// buffers and forced `unroll 1` so the accumulators stay in one register set
// (no compiler ping-pong copies) while A loads stay one k-step ahead.
__global__ void __launch_bounds__(256)
k_wmma_gemm(const float* __restrict__ A, const float* __restrict__ W,
            const float* __restrict__ rowdeg,   // null -> no scaling
            const float* __restrict__ bias,     // null -> no bias
            float* __restrict__ C, int numRowTiles) {
  __shared__ float W_lds[D * D];    // 64 KB (WGP has 320 KB LDS)

  // ---- async stage W into LDS: 16384 floats = 4096 float4, 16 per thread ----
  unsigned lds_base = (unsigned)(uintptr_t)(&W_lds[0]);  // low 32 bits = LDS byte offset
  #pragma unroll
  for (int rep = 0; rep < 16; ++rep) {
    int idx4 = rep * 256 + (int)threadIdx.x;             // float4 index
    unsigned loff = lds_base + (unsigned)idx4 * 16u;
    const float* g = W + (size_t)idx4 * 4;
    asm volatile("global_load_async_to_lds_b128 %0, %1, off"
                 :: "v"(loff), "v"(g) : "memory");
  }
  asm volatile("s_wait_asynccnt 0" ::: "memory");
  __syncthreads();

  int lane = threadIdx.x & (WAVE - 1);
  int wave = threadIdx.x >> 5;
  int tm = blockIdx.x * 8 + wave;       // 16-row tile index (wave-uniform)
  if (tm < numRowTiles) {
    int half = lane >> 4;               // 0: lanes 0-15, 1: lanes 16-31
    int l = lane & 15;

    int rowA = tm * 16 + l;
    float scale = 1.0f;
    if (rowdeg) scale = rsqrtf(fmaxf(rowdeg[rowA], 1.0f));
    const float* Arow = A + (size_t)rowA * D + half * 2;

    v8f acc[8] = {};

    // prologue: A fragment for k0 = 0
    v2f a0 = *(const v2f*)Arow;

    #pragma unroll 1
    for (int k0 = 0; k0 < D - 8; k0 += 8) {
      v2f a1 = *(const v2f*)(Arow + k0 + 4);       // prefetch step k0+4
      wmma_kstep(acc, a0, scale, W_lds, k0, half, l);
      a0 = *(const v2f*)(Arow + k0 + 8);           // prefetch step k0+8
      wmma_kstep(acc, a1, scale, W_lds, k0 + 4, half, l);
    }
    {   // tail: k-steps D-8 and D-4 (a0 already holds D-8)
      v2f a1 = *(const v2f*)(Arow + D - 4);
      wmma_kstep(acc, a0, scale, W_lds, D - 8, half, l);
      wmma_kstep(acc, a1, scale, W_lds, D - 4, half, l);
    }

    #pragma unroll
    for (int tn = 0; tn < 8; ++tn) {
      int col = tn * 16 + l;
      float bv = bias ? bias[col] : 0.0f;
      #pragma unroll
      for (int j = 0; j < 8; ++j) {
        int r = tm * 16 + j + half * 8;
        C[(size_t)r * D + col] = acc[tn][j] + bv;
      }
    }
  }
}

// ---------- 4) per-edge score = dot(q[src], k[dst]); wave32 reduce; segment max ----------
__global__ void k_score(const float* __restrict__ q, const float* __restrict__ k,
                        const int* __restrict__ src, const int* __restrict__ dst,
                        float* __restrict__ score, unsigned* __restrict__ smaxkey, int E) {
  int lane = threadIdx.x & (WAVE - 1);
  int e = (blockIdx.x * blockDim.x + threadIdx.x) >> 5;
  if (e >= E) return;
  int s = src[e], d = dst[e];
  float4 qv = ((const float4*)(q + (size_t)s * D))[lane];
  float4 kv = ((const float4*)(k + (size_t)d * D))[lane];
  float p = qv.x * kv.x + qv.y * kv.y + qv.z * kv.z + qv.w * kv.w;
  #pragma unroll
  for (int off = 16; off > 0; off >>= 1) p += __shfl_xor(p, off, WAVE);
  if (lane == 0) {
    score[e] = p;
    atomicMax(&smaxkey[d], f2key(p));
  }
}

// ---------- 5) e = exp(score - smax[dst]); denom[dst] += e ----------
__global__ void k_expdenom(const int* __restrict__ dst, const unsigned* __restrict__ smaxkey,
                           float* __restrict__ score, float* __restrict__ denom, int E) {
  int e = blockIdx.x * blockDim.x + threadIdx.x;
  if (e >= E) return;
  int d = dst[e];
  float m = key2f(smaxkey[d]);
  float ev = expf(score[e] - m);
  score[e] = ev;
  atomicAdd(&denom[d], ev);
}

// ---------- 6) outputs: [src (E), dst (E), probs (E)] ----------
__global__ void k_finalize(const int* __restrict__ src, const int* __restrict__ dst,
                           const float* __restrict__ escore, const float* __restrict__ denom,
                           float* __restrict__ out, int E) {
  int e = blockIdx.x * blockDim.x + threadIdx.x;
  if (e >= E) return;
  out[e] = (float)src[e];
  out[E + e] = (float)dst[e];
  out[2 * E + e] = escore[e] / denom[dst[e]];
}

extern "C" void kernel_launch(void* const* d_in, const int* in_sizes, int n_in,
                              void* d_out, int out_size, void* d_ws, size_t ws_size,
                              hipStream_t stream) {
  const float* x      = (const float*)d_in[0];
  const int*   src    = (const int*)d_in[1];
  const int*   dst    = (const int*)d_in[2];
  const float* W_conv = (const float*)d_in[3];
  const float* b_conv = (const float*)d_in[4];
  const float* W_q    = (const float*)d_in[5];
  const float* W_k    = (const float*)d_in[6];
  float* out = (float*)d_out;

  const int N = in_sizes[0] / D;   // 50000
  const int E = in_sizes[1];       // 800000

  // workspace layout (floats)
  float* ws       = (float*)d_ws;
  float* deg_out  = ws;                               // N
  float* deg_in   = deg_out + N;                      // N
  float* agg      = deg_in + N;                       // N*D
  float* h        = agg + (size_t)N * D;              // N*D
  float* q        = h + (size_t)N * D;                // N*D
  float* kmat     = q + (size_t)N * D;                // N*D
  float* score    = kmat + (size_t)N * D;             // E
  unsigned* smaxk = (unsigned*)(score + E);           // N
  float* denom    = (float*)(smaxk + N);              // N

  // zero-init accumulators (memset nodes are graph-capturable)
  hipMemsetAsync(deg_out, 0, sizeof(float) * 2 * N, stream);
  hipMemsetAsync(agg, 0, sizeof(float) * (size_t)N * D, stream);
  hipMemsetAsync(smaxk, 0, sizeof(unsigned) * N, stream);     // key 0 == sentinel
  hipMemsetAsync(denom, 0, sizeof(float) * N, stream);

  const int T = 256;
  k_degrees<<<(E + T - 1) / T, T, 0, stream>>>(src, dst, deg_out, deg_in, E);

  // one wave per edge
  int edgeWaveBlocks = (E * WAVE + T - 1) / T;
  k_aggregate<<<edgeWaveBlocks, T, 0, stream>>>(x, src, dst, deg_out, agg, E);

  // GEMMs: one wave per 16x128 row strip, 8 waves per block
  int rowTiles = (N + 15) / 16;               // 3125
  int gemmBlocks = (rowTiles + 7) / 8;        // 391
  k_wmma_gemm<<<gemmBlocks, T, 0, stream>>>(agg, W_conv, deg_in, b_conv, h, rowTiles);
  k_wmma_gemm<<<gemmBlocks, T, 0, stream>>>(h, W_q, nullptr, nullptr, q, rowTiles);
  k_wmma_gemm<<<gemmBlocks, T, 0, stream>>>(h, W_k, nullptr, nullptr, kmat, rowTiles);

  k_score<<<edgeWaveBlocks, T, 0, stream>>>(q, kmat, src, dst, score, smaxk, E);
  k_expdenom<<<(E + T - 1) / T, T, 0, stream>>>(dst, smaxk, score, denom, E);
  k_finalize<<<(E + T - 1) / T, T, 0, stream>>>(src, dst, score, denom, out, E);
}